// MultiPNet_76802605187633
// MI455X (gfx1250) — compile-verified
//
#include <hip/hip_runtime.h>
#include <math.h>

typedef float v2f __attribute__((ext_vector_type(2)));
typedef float v4f __attribute__((ext_vector_type(4)));
typedef float v8f __attribute__((ext_vector_type(8)));

#define B_TOT   16384
#define K_TOT   10000
#define FEAT    128
#define PREF    2
#define NEXP    4
#define ACT     8
#define SLOPE   0.01f
#define PIECE   0.39269908169872414f   /* pi/2/4 */

// ---------------------------------------------------------------------------
// CDNA5 async global->LDS copy (ASYNCcnt-tracked), ISA §10 / §15.18.3.
// LDS destination address = low 32 bits of the generic pointer (LDS aperture).
// ---------------------------------------------------------------------------
__device__ __forceinline__ void async_b128(void* lds_ptr, const void* gptr) {
  unsigned int       loff = (unsigned int)(uintptr_t)lds_ptr;
  unsigned long long gad  = (unsigned long long)(uintptr_t)gptr;
  asm volatile("global_load_async_to_lds_b128 %0, %1, off"
               :: "v"(loff), "v"(gad) : "memory");
}

// ---------------------------------------------------------------------------
// Kernel 1: x = leaky_relu(obs @ Wf + bf)   [16384,10000] x [10000,128]
// 64 rows x 128 cols per WG, 8 waves, fp32 WMMA 16x16x4.
// obs tiles: double-buffered async global->LDS.  Wf tiles: staged transposed
// so B fragments are single ds_load_b64 (no repack movs).
// ---------------------------------------------------------------------------
#define R1    64
#define KT1   32
#define NFULL 312          /* 312*32 = 9984; tail = 16 */

__global__ __launch_bounds__(256) void feat_gemm_kernel(
    const float* __restrict__ obs, const float* __restrict__ Wf,
    const float* __restrict__ bf, float* __restrict__ feat)
{
  __shared__ __align__(16) float As [2][R1][KT1 + 4];   // stride 36 (144B rows)
  __shared__ __align__(16) float BsT[2][FEAT][KT1 + 2]; // transposed, stride 34

  const int tid  = threadIdx.x;
  const int lane = tid & 31;
  const int wave = tid >> 5;
  const int r    = lane & 15;          // M (A) / N (B) index
  const int kof  = (lane >> 4) * 2;    // K pair select per lane half
  const int hi   = (lane >> 4) * 8;    // C/D row offset per lane half
  const int rowBase = (wave >> 1) * 16;
  const int colBase = (wave & 1) * 64;
  const int wgRow   = blockIdx.x * R1;

  v8f acc[4] = {};

  // --- tile stagers -------------------------------------------------------
  auto stageA = [&](int bf_, int kt) {               // async: 64x32 f32
    const int k0 = kt * KT1;
    #pragma unroll
    for (int it = 0; it < 2; ++it) {                 // 512 b128 chunks / 256 thr
      const int i  = tid + it * 256;
      const int rr = i >> 3;
      const int cc = (i & 7) * 4;
      async_b128(&As[bf_][rr][cc],
                 obs + (size_t)(wgRow + rr) * K_TOT + k0 + cc);
    }
  };
  auto stageB = [&](int bf_, int kt) {               // transposed store
    const int k0 = kt * KT1;
    #pragma unroll
    for (int it = 0; it < 4; ++it) {                 // 1024 v4f / 256 thr
      const int i  = tid + it * 256;
      const int rr = i & 31;                         // k within tile
      const int cc = (i >> 5) * 4;                   // col base
      v4f v = *(const v4f*)(Wf + (size_t)(k0 + rr) * FEAT + cc);
      BsT[bf_][cc + 0][rr] = v[0];
      BsT[bf_][cc + 1][rr] = v[1];
      BsT[bf_][cc + 2][rr] = v[2];
      BsT[bf_][cc + 3][rr] = v[3];
    }
  };
  auto compute = [&](int bf_, int nk) {              // nk k-steps of 4
    for (int kk = 0; kk < nk * 4; kk += 4) {
      const v2f a = *(const v2f*)&As[bf_][rowBase + r][kk + kof];
      #pragma unroll
      for (int t = 0; t < 4; ++t) {
        const v2f b = *(const v2f*)&BsT[bf_][colBase + t * 16 + r][kk + kof];
        acc[t] = __builtin_amdgcn_wmma_f32_16x16x4_f32(
            false, a, false, b, (short)0, acc[t], false, false);
      }
    }
  };

  // --- main loop: double-buffered -----------------------------------------
  stageA(0, 0);
  stageB(0, 0);
  for (int kt = 0; kt < NFULL; ++kt) {
    const int cur = kt & 1;
    if (kt + 1 < NFULL) {
      stageA(cur ^ 1, kt + 1);
      stageB(cur ^ 1, kt + 1);
      asm volatile("s_wait_asynccnt 0x2" ::: "memory");  // cur tile landed
    } else {
      asm volatile("s_wait_asynccnt 0x0" ::: "memory");
    }
    __syncthreads();
    compute(cur, KT1 / 4);
    __syncthreads();
  }

  // --- tail tile: k = 9984..9999 (exactly 16 wide), synchronous -----------
  {
    const int k0 = NFULL * KT1;
    for (int i = tid; i < R1 * 4; i += 256) {        // 64 rows x 4 v4f
      const int rr = i >> 2;
      const int cc = (i & 3) * 4;
      v4f v = *(const v4f*)(obs + (size_t)(wgRow + rr) * K_TOT + k0 + cc);
      *(v4f*)&As[0][rr][cc] = v;
    }
    for (int i = tid; i < 16 * 32; i += 256) {       // 16 k x 32 col-chunks
      const int rr = i & 15;
      const int cc = (i >> 4) * 4;
      v4f v = *(const v4f*)(Wf + (size_t)(k0 + rr) * FEAT + cc);
      BsT[0][cc + 0][rr] = v[0];
      BsT[0][cc + 1][rr] = v[1];
      BsT[0][cc + 2][rr] = v[2];
      BsT[0][cc + 3][rr] = v[3];
    }
    __syncthreads();
    compute(0, 4);
  }

  // --- fused bias + LeakyReLU epilogue ------------------------------------
  #pragma unroll
  for (int t = 0; t < 4; ++t) {
    const int col  = colBase + t * 16 + r;
    const float bb = bf[col];
    #pragma unroll
    for (int i = 0; i < 8; ++i) {
      const int row = wgRow + rowBase + hi + i;
      float v = acc[t][i] + bb;
      feat[(size_t)row * FEAT + col] = (v >= 0.f) ? v : SLOPE * v;
    }
  }
}

// ---------------------------------------------------------------------------
// Kernel 2: expert MLPs + value head, 32 samples per workgroup (<3% of work).
// ---------------------------------------------------------------------------
#define R2 32

__global__ __launch_bounds__(256) void heads_kernel(
    const float* __restrict__ feat, const float* __restrict__ pref,
    const float* __restrict__ Wa1, const float* __restrict__ ba1,
    const float* __restrict__ Wa2, const float* __restrict__ ba2,
    const float* __restrict__ Wc1, const float* __restrict__ bc1,
    const float* __restrict__ Wc2, const float* __restrict__ bc2,
    float* __restrict__ out_pol, float* __restrict__ out_val)
{
  __shared__ __align__(16) float Xs[R2][132];   // [feat | p0 p1 | 0 0]
  __shared__ __align__(16) float Hs[R2][132];   // hidden (cols 0..127 used)
  __shared__ __align__(16) float Ws[32][128];   // first-layer K-tile
  __shared__ __align__(16) float W2[128][16];   // second-layer, N padded to 16
  __shared__ int sIdx[R2];

  const int tid  = threadIdx.x;
  const int lane = tid & 31;
  const int wave = tid >> 5;
  const int r    = lane & 15;
  const int kof  = (lane >> 4) * 2;
  const int hi   = (lane >> 4) * 8;
  const int wgRow = blockIdx.x * R2;

  for (int i = tid; i < R2 * 32; i += 256) {
    const int rr = i >> 5;
    const int cc = (i & 31) * 4;
    v4f v = *(const v4f*)(feat + (size_t)(wgRow + rr) * FEAT + cc);
    *(v4f*)&Xs[rr][cc] = v;
  }
  if (tid < R2) {
    const float p0 = pref[(wgRow + tid) * 2 + 0];
    const float p1 = pref[(wgRow + tid) * 2 + 1];
    Xs[tid][128] = p0; Xs[tid][129] = p1;
    Xs[tid][130] = 0.f; Xs[tid][131] = 0.f;
    const float theta = atanf(p1 / (p0 + 0.01f));
    int e = (int)floorf(theta / PIECE);
    sIdx[tid] = e < 0 ? 0 : (e > NEXP - 1 ? NEXP - 1 : e);
  }
  __syncthreads();

  const int rowT = (wave >> 2) * 16;   // 0 or 16
  const int colB = (wave & 3) * 32;    // 0,32,64,96

  for (int m = 0; m < 5; ++m) {
    const float* W1 = (m < NEXP) ? (Wa1 + (size_t)m * (FEAT + PREF) * FEAT) : Wc1;
    const float* b1 = (m < NEXP) ? (ba1 + m * FEAT) : bc1;

    v8f acc[2] = {};
    for (int kt = 0; kt < 4; ++kt) {
      const int k0 = kt * 32;
      for (int i = tid; i < 32 * 32; i += 256) {
        const int rr = i >> 5;
        const int cc = (i & 31) * 4;
        *(v4f*)&Ws[rr][cc] = *(const v4f*)(W1 + (size_t)(k0 + rr) * FEAT + cc);
      }
      __syncthreads();
      #pragma unroll
      for (int kk = 0; kk < 32; kk += 4) {
        v2f a;
        a.x = Xs[rowT + r][k0 + kk + kof];
        a.y = Xs[rowT + r][k0 + kk + kof + 1];
        #pragma unroll
        for (int t = 0; t < 2; ++t) {
          v2f b;
          b.x = Ws[kk + kof][colB + t * 16 + r];
          b.y = Ws[kk + kof + 1][colB + t * 16 + r];
          acc[t] = __builtin_amdgcn_wmma_f32_16x16x4_f32(
              false, a, false, b, (short)0, acc[t], false, false);
        }
      }
      __syncthreads();
    }
    // K tail: rows 128,129 (preference), 130,131 zero-padded
    for (int i = tid; i < 4 * 32; i += 256) {
      const int rr = i >> 5;
      const int cc = (i & 31) * 4;
      v4f v = {0.f, 0.f, 0.f, 0.f};
      if (128 + rr < FEAT + PREF)
        v = *(const v4f*)(W1 + (size_t)(128 + rr) * FEAT + cc);
      *(v4f*)&Ws[rr][cc] = v;
    }
    __syncthreads();
    {
      v2f a;
      a.x = Xs[rowT + r][128 + kof];
      a.y = Xs[rowT + r][128 + kof + 1];
      #pragma unroll
      for (int t = 0; t < 2; ++t) {
        v2f b;
        b.x = Ws[kof][colB + t * 16 + r];
        b.y = Ws[kof + 1][colB + t * 16 + r];
        acc[t] = __builtin_amdgcn_wmma_f32_16x16x4_f32(
            false, a, false, b, (short)0, acc[t], false, false);
      }
    }

    #pragma unroll
    for (int t = 0; t < 2; ++t) {
      const int col  = colB + t * 16 + r;
      const float bb = b1[col];
      #pragma unroll
      for (int i = 0; i < 8; ++i) {
        float v = acc[t][i] + bb;
        Hs[rowT + hi + i][col] = (v >= 0.f) ? v : SLOPE * v;
      }
    }
    const float* Wsec = (m < NEXP) ? (Wa2 + (size_t)m * FEAT * ACT) : Wc2;
    const int nc2 = (m < NEXP) ? ACT : PREF;
    for (int i = tid; i < 128 * 16; i += 256) {
      const int rr = i >> 4;
      const int cc = i & 15;
      W2[rr][cc] = (cc < nc2) ? Wsec[rr * nc2 + cc] : 0.f;
    }
    __syncthreads();

    if (wave < 2) {               // wave-uniform: EXEC all-ones inside
      const int rT = wave * 16;
      v8f a2 = {};
      #pragma unroll
      for (int kk = 0; kk < FEAT; kk += 4) {
        v2f a;
        a.x = Hs[rT + r][kk + kof];
        a.y = Hs[rT + r][kk + kof + 1];
        v2f b;
        b.x = W2[kk + kof][r];
        b.y = W2[kk + kof + 1][r];
        a2 = __builtin_amdgcn_wmma_f32_16x16x4_f32(
            false, a, false, b, (short)0, a2, false, false);
      }
      if (m < NEXP) {
        if (r < ACT) {
          const float bb = ba2[m * ACT + r];
          #pragma unroll
          for (int i = 0; i < 8; ++i) {
            const int lr = rT + hi + i;
            if (sIdx[lr] == m)
              out_pol[(size_t)(wgRow + lr) * ACT + r] = a2[i] + bb;
          }
        }
      } else {
        if (r < PREF) {
          const float bb = bc2[r];
          #pragma unroll
          for (int i = 0; i < 8; ++i) {
            const int lr = rT + hi + i;
            out_val[(size_t)(wgRow + lr) * PREF + r] = a2[i] + bb;
          }
        }
      }
    }
    __syncthreads();
  }
}

// ---------------------------------------------------------------------------
extern "C" void kernel_launch(void* const* d_in, const int* in_sizes, int n_in,
                              void* d_out, int out_size, void* d_ws, size_t ws_size,
                              hipStream_t stream) {
  (void)in_sizes; (void)n_in; (void)out_size; (void)ws_size;
  const float* obs  = (const float*)d_in[0];
  const float* pref = (const float*)d_in[1];
  const float* Wf   = (const float*)d_in[2];
  const float* bf   = (const float*)d_in[3];
  const float* Wa1  = (const float*)d_in[4];
  const float* ba1  = (const float*)d_in[5];
  const float* Wa2  = (const float*)d_in[6];
  const float* ba2  = (const float*)d_in[7];
  const float* Wc1  = (const float*)d_in[8];
  const float* bc1  = (const float*)d_in[9];
  const float* Wc2  = (const float*)d_in[10];
  const float* bc2  = (const float*)d_in[11];

  float* feat    = (float*)d_ws;                      // 16384*128 f32 = 8 MB
  float* out_pol = (float*)d_out;                     // [B, 8]
  float* out_val = out_pol + (size_t)B_TOT * ACT;     // [B, 2]

  feat_gemm_kernel<<<B_TOT / R1, 256, 0, stream>>>(obs, Wf, bf, feat);
  heads_kernel<<<B_TOT / R2, 256, 0, stream>>>(feat, pref, Wa1, ba1, Wa2, ba2,
                                               Wc1, bc1, Wc2, bc2,
                                               out_pol, out_val);
}